// Encoder_22943715295569
// MI455X (gfx1250) — compile-verified
//
#include <hip/hip_runtime.h>

typedef __attribute__((ext_vector_type(2))) float v2f;
typedef __attribute__((ext_vector_type(8))) float v8f;

constexpr int kNodes = 20000;
constexpr int kEdges = 320000;
constexpr int kInDim = 2000;
constexpr int kHDim  = 256;
constexpr int kZDim  = 128;

__device__ __forceinline__ float gelu_exact(float x) {
    // torch nn.GELU default: 0.5*x*(1+erf(x/sqrt(2)))
    return 0.5f * x * (1.0f + erff(x * 0.7071067811865476f));
}

// C[M,N] = act(A[M,K] @ B[K,N] + bias), fp32 WMMA 16x16x4.
// One wave computes a 16-row strip across the full N width (NT = N/16 tiles).
template<int NT, bool HAS_BIAS, bool APPLY_GELU>
__global__ __launch_bounds__(256) void gemm_wmma_f32(
    const float* __restrict__ A, const float* __restrict__ B,
    const float* __restrict__ bias, float* __restrict__ C, int M, int K)
{
    constexpr int N = NT * 16;
    const int wave = threadIdx.x >> 5;
    const int lane = threadIdx.x & 31;
    const int half = lane >> 4;   // 0: lanes 0-15, 1: lanes 16-31
    const int l15  = lane & 15;
    const int mt   = blockIdx.x * 8 + wave;
    if (mt * 16 >= M) return;     // wave-uniform: EXEC stays all-1s for WMMA

    // A fragment (16x4 f32): lane l<16 holds A[l][k0+0..1], lane l+16 holds A[l][k0+2..3]
    const float* Arow = A + (size_t)(mt * 16 + l15) * K + (half << 1);

    v8f acc[NT] = {};

    for (int k0 = 0; k0 < K; k0 += 4) {
        const float2 a2 = *(const float2*)(Arow + k0);
        v2f a; a[0] = a2.x; a[1] = a2.y;
        // B fragment (4x16): lane = column n, v0 holds B[k0+2*half][n], v1 holds B[k0+2*half+1][n]
        const float* B0 = B + (size_t)(k0 + (half << 1)) * N + l15;
        const float* B1 = B0 + N;
#pragma unroll
        for (int t = 0; t < NT; ++t) {
            v2f b; b[0] = B0[t * 16]; b[1] = B1[t * 16];
            acc[t] = __builtin_amdgcn_wmma_f32_16x16x4_f32(
                false, a, false, b, (short)0, acc[t], false, false);
        }
    }

    // D layout (16x16 f32): lane = column n (l15), VGPR v + 8*half = row
#pragma unroll
    for (int t = 0; t < NT; ++t) {
        const int n = t * 16 + l15;
        const float bv = HAS_BIAS ? bias[n] : 0.0f;
#pragma unroll
        for (int v = 0; v < 8; ++v) {
            const int m = mt * 16 + v + (half << 3);
            float r = acc[t][v] + bv;
            if (APPLY_GELU) r = gelu_exact(r);
            C[(size_t)m * N + n] = r;
        }
    }
}

__global__ __launch_bounds__(256) void k_deg_init(float* __restrict__ deg) {
    int i = blockIdx.x * blockDim.x + threadIdx.x;
    if (i < kNodes) deg[i] = 1.0f;  // self-loop weight
}

__global__ __launch_bounds__(256) void k_deg_accum(
    const int* __restrict__ col, const float* __restrict__ ew, float* __restrict__ deg) {
    int e = blockIdx.x * blockDim.x + threadIdx.x;
    if (e < kEdges)
        __hip_atomic_fetch_add(&deg[col[e]], ew[e], __ATOMIC_RELAXED, __HIP_MEMORY_SCOPE_AGENT);
}

__global__ __launch_bounds__(256) void k_inv_sqrt(float* __restrict__ deg) {
    int i = blockIdx.x * blockDim.x + threadIdx.x;
    if (i < kNodes) {
        float d = deg[i];
        deg[i] = (d > 0.0f) ? rsqrtf(d) : 0.0f;
    }
}

// out[i][f] = xw[i][f] * dis[i]^2   (self-loop contribution; also zero-initializes out)
__global__ __launch_bounds__(256) void k_selfloop_init(
    const float* __restrict__ xw, const float* __restrict__ dis, float* __restrict__ out) {
    int idx = blockIdx.x * blockDim.x + threadIdx.x;
    if (idx < kNodes * kZDim) {
        int i = idx >> 7;
        float s = dis[i];
        out[idx] = xw[idx] * s * s;
    }
}

// out[col[e]][f] += xw[row[e]][f] * dis[row]*ew*dis[col]
__global__ __launch_bounds__(256) void k_edge_scatter(
    const int* __restrict__ row, const int* __restrict__ col,
    const float* __restrict__ ew, const float* __restrict__ dis,
    const float* __restrict__ xw, float* __restrict__ out) {
    int idx = blockIdx.x * blockDim.x + threadIdx.x;   // kEdges*128 = 40.96M < 2^31
    if (idx < kEdges * kZDim) {
        int e = idx >> 7;
        int f = idx & (kZDim - 1);
        int r = row[e], c = col[e];
        float nrm = dis[r] * ew[e] * dis[c];
        __hip_atomic_fetch_add(&out[(size_t)c * kZDim + f],
                               xw[(size_t)r * kZDim + f] * nrm,
                               __ATOMIC_RELAXED, __HIP_MEMORY_SCOPE_AGENT);
    }
}

__global__ __launch_bounds__(256) void k_bias_gelu(
    float* __restrict__ x, const float* __restrict__ b) {
    int idx = blockIdx.x * blockDim.x + threadIdx.x;
    if (idx < kNodes * kZDim)
        x[idx] = gelu_exact(x[idx] + b[idx & (kZDim - 1)]);
}

extern "C" void kernel_launch(void* const* d_in, const int* in_sizes, int n_in,
                              void* d_out, int out_size, void* d_ws, size_t ws_size,
                              hipStream_t stream) {
    (void)in_sizes; (void)n_in; (void)out_size; (void)ws_size;
    const float* x    = (const float*)d_in[0];
    const int*   ei   = (const int*)d_in[1];
    const float* ew   = (const float*)d_in[2];
    const float* W1   = (const float*)d_in[3];
    const float* b1   = (const float*)d_in[4];
    const float* W2   = (const float*)d_in[5];
    const float* b2   = (const float*)d_in[6];
    const float* gW1  = (const float*)d_in[7];
    const float* gb1  = (const float*)d_in[8];
    const float* gW2  = (const float*)d_in[9];
    const float* gb2  = (const float*)d_in[10];
    const float* lW   = (const float*)d_in[11];
    const float* lb   = (const float*)d_in[12];
    const float* genW = (const float*)d_in[13];
    const float* genb = (const float*)d_in[14];
    const int* row = ei;
    const int* col = ei + kEdges;
    float* out = (float*)d_out;

    // Workspace layout with reuse (~51.3 MB total):
    float* ws = (float*)d_ws;
    const size_t SZ_H = (size_t)kNodes * kHDim;  // 5.12M floats
    const size_t SZ_Z = (size_t)kNodes * kZDim;  // 2.56M floats
    float* buf0 = ws;                       // h1 -> xw1 -> xw2 -> z3
    float* buf1 = ws + SZ_H;                // h2 -> g2
    float* buf2 = ws + 2 * SZ_H;            // g1
    float* dis  = ws + 2 * SZ_H + SZ_Z;     // deg then dis, kNodes floats

    const int TPB = 256;
    const int mblocks = (kNodes / 16 + 7) / 8;                // 157 (8 waves/block)
    const int nzb = (kNodes * kZDim + TPB - 1) / TPB;         // 10000
    const int neb = (kEdges * kZDim + TPB - 1) / TPB;         // 160000
    const int nb  = (kNodes + TPB - 1) / TPB;
    const int eb  = (kEdges + TPB - 1) / TPB;

    // Dense MLP: h1 = gelu(x@W1+b1); h2 = gelu(h1@W2+b2)
    gemm_wmma_f32<16, true, true><<<mblocks, TPB, 0, stream>>>(x,    W1, b1, buf0, kNodes, kInDim);
    gemm_wmma_f32<16, true, true><<<mblocks, TPB, 0, stream>>>(buf0, W2, b2, buf1, kNodes, kHDim);

    // Symmetric degree normalization (deg includes self-loop weight 1)
    k_deg_init <<<nb, TPB, 0, stream>>>(dis);
    k_deg_accum<<<eb, TPB, 0, stream>>>(col, ew, dis);
    k_inv_sqrt <<<nb, TPB, 0, stream>>>(dis);

    // GCN layer 1: xw1 = h2@gW1; g1 = gelu(scatter(xw1) + gb1)
    gemm_wmma_f32<8, false, false><<<mblocks, TPB, 0, stream>>>(buf1, gW1, nullptr, buf0, kNodes, kHDim);
    k_selfloop_init<<<nzb, TPB, 0, stream>>>(buf0, dis, buf2);
    k_edge_scatter <<<neb, TPB, 0, stream>>>(row, col, ew, dis, buf0, buf2);
    k_bias_gelu    <<<nzb, TPB, 0, stream>>>(buf2, gb1);

    // GCN layer 2: xw2 = g1@gW2; g2 = gelu(scatter(xw2) + gb2)
    gemm_wmma_f32<8, false, false><<<mblocks, TPB, 0, stream>>>(buf2, gW2, nullptr, buf0, kNodes, kZDim);
    k_selfloop_init<<<nzb, TPB, 0, stream>>>(buf0, dis, buf1);
    k_edge_scatter <<<neb, TPB, 0, stream>>>(row, col, ew, dis, buf0, buf1);
    k_bias_gelu    <<<nzb, TPB, 0, stream>>>(buf1, gb2);

    // Heads: z3 = g2@lW + lb; out = z3@genW + genb
    gemm_wmma_f32<8, true, false><<<mblocks, TPB, 0, stream>>>(buf1, lW,   lb,   buf0, kNodes, kZDim);
    gemm_wmma_f32<8, true, false><<<mblocks, TPB, 0, stream>>>(buf0, genW, genb, out,  kNodes, kZDim);
}